// AffineTransformStripe_66468913873022
// MI455X (gfx1250) — compile-verified
//
#include <hip/hip_runtime.h>

typedef float v2f __attribute__((ext_vector_type(2)));
typedef float v8f __attribute__((ext_vector_type(8)));
typedef float f4  __attribute__((ext_vector_type(4)));

#define LN100 4.6051701859880913680359829093687f

// ---------------------------------------------------------------------------
// Kernel A: CPB-MLP second layer via V_WMMA_F32_16X16X4_F32 (exact fp32).
//   h[t][k] = relu(table[t]·w1[:,k] + b1[k]) computed on the fly,
//   bias_table = h @ w2   (T=529 padded to 544 rows, H=6 padded to 16 cols)
// One wave per 16-row output tile; K=512 consumed 4 at a time.
// A 16x4 layout: lanes 0-15 hold K={k0,k0+1}, lanes 16-31 hold K={k0+2,k0+3}.
// B 4x16 layout: VGPR0 rows {k0 | k0+2}, VGPR1 rows {k0+1 | k0+3}, N across lanes.
// ---------------------------------------------------------------------------
__global__ __launch_bounds__(32)
void cpb_mlp_wmma(const float* __restrict__ table,   // [529,2]
                  const float* __restrict__ w1,      // [2,512]
                  const float* __restrict__ b1,      // [512]
                  const float* __restrict__ w2,      // [512,6]
                  float* __restrict__ tableWs)       // [544,16]
{
    const int lane = threadIdx.x;
    const int lm   = lane & 15;
    const int half = lane >> 4;
    const int rowbase = blockIdx.x * 16;
    const int row  = rowbase + lm;

    // Branch-free pad-row masking (keeps EXEC all-ones for WMMA)
    const float rvalid = (row < 529) ? 1.0f : 0.0f;
    const int   rc     = (row < 529) ? row : 528;
    const float tx = table[rc * 2 + 0] * rvalid;
    const float ty = table[rc * 2 + 1] * rvalid;

    const float nmask = (lm < 6) ? 1.0f : 0.0f;
    const int   nc    = (lm < 6) ? lm : 0;

    v8f acc = {0.f, 0.f, 0.f, 0.f, 0.f, 0.f, 0.f, 0.f};

    for (int k0 = 0; k0 < 512; k0 += 4) {
        const int j0 = k0 + 2 * half;
        // A operand: two hidden activations for this lane's K slots
        float h0 = fmaxf(fmaf(tx, w1[j0],     fmaf(ty, w1[512 + j0],     b1[j0])),     0.0f);
        float h1 = fmaxf(fmaf(tx, w1[j0 + 1], fmaf(ty, w1[512 + j0 + 1], b1[j0 + 1])), 0.0f);
        v2f a; a.x = h0; a.y = h1;
        // B operand: w2 rows j0, j0+1 striped across lanes (cols >= 6 zeroed)
        v2f b;
        b.x = w2[j0 * 6 + nc]       * nmask;
        b.y = w2[(j0 + 1) * 6 + nc] * nmask;
        acc = __builtin_amdgcn_wmma_f32_16x16x4_f32(
                  /*neg_a=*/false, a, /*neg_b=*/false, b,
                  /*c_mod=*/(short)0, acc, /*reuse_a=*/false, /*reuse_b=*/false);
    }

    // D layout: VGPR r -> M=r (lanes 0-15) / M=8+r (lanes 16-31), N = lane&15
#pragma unroll
    for (int r = 0; r < 8; ++r) {
        int orow = rowbase + half * 8 + r;
        tableWs[orow * 16 + lm] = acc[r];
    }
}

// ---------------------------------------------------------------------------
// Kernel B: relative-position gather + 16*sigmoid into a [6,256,64] grid,
// plus the 6 per-head logit scales exp(min(ls, ln 100)).
// ---------------------------------------------------------------------------
__global__ __launch_bounds__(256)
void bias_gather(const float* __restrict__ tableWs,     // [544,16]
                 const int*   __restrict__ index,       // [256*64]
                 const float* __restrict__ logit_scale, // [6]
                 float* __restrict__ biasgrid,          // [6*16384]
                 float* __restrict__ scales)            // [6]
{
    int tid = blockIdx.x * 256 + threadIdx.x;
    if (tid < 6 * 16384) {
        int h   = tid >> 14;        // 16384 = 2^14
        int rem = tid & 16383;
        int idx = index[rem];
        float v = tableWs[idx * 16 + h];
        biasgrid[tid] = 16.0f / (1.0f + __expf(-v));
    }
    if (tid < 6) {
        scales[tid] = __expf(fminf(logit_scale[tid], LN100));
    }
}

// ---------------------------------------------------------------------------
// Kernel C: bandwidth-bound stream. One float4 per thread; attn/out are
// non-temporal (805 MB stream > 192 MB L2); mask (16.8 MB) and biasgrid
// (0.4 MB) stay L2-resident and are re-read with regular temporal policy.
// h / b_ / w are uniform per 256-thread block (blocks never straddle a
// 4096-float4 [n1,n2] plane), so those reads scalarize.
// ---------------------------------------------------------------------------
__global__ __launch_bounds__(256)
void apply_all(const f4* __restrict__ attn4,      // [1024*6*256*16]
               const f4* __restrict__ biasgrid4,  // [6*4096]
               const f4* __restrict__ mask4,      // [256*4096]
               const float* __restrict__ scales,  // [6]
               f4* __restrict__ out4)
{
    int i   = blockIdx.x * 256 + threadIdx.x;   // < 25,165,824
    int n2v = i & 15;
    int n1  = (i >> 4) & 255;
    int bh  = i >> 12;          // block-uniform
    int h   = bh % 6;
    int b_  = bh / 6;
    int w   = b_ & 255;

    float s = scales[h];
    f4 a  = __builtin_nontemporal_load(attn4 + i);
    f4 bb = biasgrid4[h * 4096 + n1 * 16 + n2v];
    f4 m  = mask4[w * 4096 + n1 * 16 + n2v];
    f4 o  = a * s + (bb + m);
    __builtin_nontemporal_store(o, out4 + i);
}

// ---------------------------------------------------------------------------
extern "C" void kernel_launch(void* const* d_in, const int* in_sizes, int n_in,
                              void* d_out, int out_size, void* d_ws, size_t ws_size,
                              hipStream_t stream) {
    const float* attn        = (const float*)d_in[0];
    const float* logit_scale = (const float*)d_in[1];
    const float* w1          = (const float*)d_in[2];
    const float* b1          = (const float*)d_in[3];
    const float* w2          = (const float*)d_in[4];
    const float* table       = (const float*)d_in[5];
    const int*   index       = (const int*)d_in[6];
    const float* mask        = (const float*)d_in[7];
    float* out = (float*)d_out;

    // Workspace layout (floats): [0,8704) tableWs, [8704,8710) scales,
    // [8720, 8720+98304) biasgrid (16B aligned). Total ~428 KB.
    float* tableWs  = (float*)d_ws;
    float* scales   = tableWs + 8704;
    float* biasgrid = tableWs + 8720;

    cpb_mlp_wmma<<<34, 32, 0, stream>>>(table, w1, b1, w2, tableWs);
    bias_gather<<<(6 * 16384 + 255) / 256, 256, 0, stream>>>(
        tableWs, index, logit_scale, biasgrid, scales);

    const int total4 = 1024 * 6 * 256 * 16;   // 25,165,824 float4s
    apply_all<<<total4 / 256, 256, 0, stream>>>(
        (const f4*)attn, (const f4*)biasgrid, (const f4*)mask, scales, (f4*)out);
}